// OneToOneDeepNet_12378095747556
// MI455X (gfx1250) — compile-verified
//
#include <hip/hip_runtime.h>

// ---------------------------------------------------------------------------
// OneToOneDeepNet on MI455X (gfx1250).
// HBM-bound streaming kernel: 8 B traffic/element, ~17 VALU ops/element.
//  - b128 non-temporal loads/stores (512 MB stream >> 192 MB L2)
//  - gfx1250 v_tanh_f32 (single TRANS op per tanh)
//  - 16 rows fully unrolled: all 16 NT loads issued as one clause up front
//    (8 KB in flight per wave) before the tanh chains consume them.
// ---------------------------------------------------------------------------

#define D_OUT_C 1024
#define ROWS_PER_BLOCK 16

typedef float f4 __attribute__((ext_vector_type(4)));

__device__ __forceinline__ float dev_tanh(float x) {
#if __has_builtin(__builtin_amdgcn_tanhf)
  return __builtin_amdgcn_tanhf(x);   // V_TANH_F32
#else
  return tanhf(x);
#endif
}

__global__ __launch_bounds__(256) void onetoone_deepnet_kernel(
    const float* __restrict__ x,
    const float* __restrict__ W,
    const float* __restrict__ B,
    float* __restrict__ out,
    int batch)
{
  // Thread t owns channel slice [4t, 4t+4) of every row this block touches,
  // so the 8 per-channel weight/bias float4s live in registers throughout.
  const int c = (int)threadIdx.x * 4;

  const f4 w0 = *(const f4*)(W + 0 * D_OUT_C + c);
  const f4 w1 = *(const f4*)(W + 1 * D_OUT_C + c);
  const f4 w2 = *(const f4*)(W + 2 * D_OUT_C + c);
  const f4 w3 = *(const f4*)(W + 3 * D_OUT_C + c);
  const f4 b0 = *(const f4*)(B + 0 * D_OUT_C + c);
  const f4 b1 = *(const f4*)(B + 1 * D_OUT_C + c);
  const f4 b2 = *(const f4*)(B + 2 * D_OUT_C + c);
  const f4 b3 = *(const f4*)(B + 3 * D_OUT_C + c);

  const int r0 = (int)blockIdx.x * ROWS_PER_BLOCK;
  const float* px = x   + (size_t)r0 * D_OUT_C + c;
  float*       po = out + (size_t)r0 * D_OUT_C + c;

  if (r0 + ROWS_PER_BLOCK <= batch) {
    // ---- fast path: full tile, compile-time trip counts ----
    f4 v[ROWS_PER_BLOCK];

    // Issue all 16 NT b128 loads first (immediate offsets i*4096 fit the
    // 24-bit IOFFSET) -> one load clause, 8 KB in flight per wave.
#pragma unroll
    for (int i = 0; i < ROWS_PER_BLOCK; ++i)
      v[i] = __builtin_nontemporal_load((const f4*)(px + i * D_OUT_C));

#pragma unroll
    for (int i = 0; i < ROWS_PER_BLOCK; ++i) {
      f4 t = v[i];
#pragma unroll
      for (int j = 0; j < 4; ++j) {
        float h = t[j];
        h = dev_tanh(__builtin_fmaf(h, w0[j], b0[j]));
        h = dev_tanh(__builtin_fmaf(h, w1[j], b1[j]));
        h = dev_tanh(__builtin_fmaf(h, w2[j], b2[j]));
        t[j] = __builtin_fmaf(h, w3[j], b3[j]);
      }
      __builtin_nontemporal_store(t, (f4*)(po + i * D_OUT_C));
    }
  } else {
    // ---- tail path: generic row loop ----
    for (int r = r0; r < batch; ++r) {
      f4 t = __builtin_nontemporal_load((const f4*)px);
#pragma unroll
      for (int j = 0; j < 4; ++j) {
        float h = t[j];
        h = dev_tanh(__builtin_fmaf(h, w0[j], b0[j]));
        h = dev_tanh(__builtin_fmaf(h, w1[j], b1[j]));
        h = dev_tanh(__builtin_fmaf(h, w2[j], b2[j]));
        t[j] = __builtin_fmaf(h, w3[j], b3[j]);
      }
      __builtin_nontemporal_store(t, (f4*)po);
      px += D_OUT_C;
      po += D_OUT_C;
    }
  }
}

extern "C" void kernel_launch(void* const* d_in, const int* in_sizes, int n_in,
                              void* d_out, int out_size, void* d_ws, size_t ws_size,
                              hipStream_t stream) {
  (void)n_in; (void)out_size; (void)d_ws; (void)ws_size;

  const float* x = (const float*)d_in[0];   // [BATCH, 1024] f32
  const float* W = (const float*)d_in[1];   // [4, 1024] f32
  const float* B = (const float*)d_in[2];   // [4, 1024] f32
  float* out = (float*)d_out;               // [BATCH, 1024] f32

  const int batch = in_sizes[0] / D_OUT_C;  // 65536
  const int grid = (batch + ROWS_PER_BLOCK - 1) / ROWS_PER_BLOCK;  // 4096

  onetoone_deepnet_kernel<<<grid, 256, 0, stream>>>(x, W, B, out, batch);
}